// PlanarConsistencyLoss_54382875902439
// MI455X (gfx1250) — compile-verified
//
#include <hip/hip_runtime.h>
#include <hip/hip_bf16.h>
#include <stdint.h>

// ---------------------------------------------------------------------------
// PlanarConsistencyLoss for MI455X (gfx1250).
// HBM-bound gather+reduction: 64 MB traffic -> ~2.8us floor at 23.3 TB/s.
// CDNA5 paths used: TDM tensor_load_to_lds for plane-table staging,
// v_wmma_f32_16x16x4_f32 for the exact-f32 block reduction.
// ---------------------------------------------------------------------------

typedef float v2f  __attribute__((ext_vector_type(2)));
typedef float v8f  __attribute__((ext_vector_type(8)));
typedef unsigned int u32x4 __attribute__((ext_vector_type(4)));
typedef int   i32x8 __attribute__((ext_vector_type(8)));
typedef int   i32x4 __attribute__((ext_vector_type(4)));

#define BLOCK 256
#define NPLANES 64

__device__ __forceinline__ unsigned lds_offset_of(const void* p) {
  // Flat LDS aperture: low 32 bits of the generic address are the LDS byte offset.
  return (unsigned)(uintptr_t)p;
}

#if __has_builtin(__builtin_amdgcn_tensor_load_to_lds)
// 1-D contiguous f32 copy, global -> LDS, via the Tensor Data Mover.
// D# per CDNA5 ISA ch.8: group0 {count=1, lds_addr, global_addr, type=2},
// group1 {data_size=4B, tensor_dim0=tile_dim0=nelem, tensor_dim1=tile_dim1=1}.
__device__ __forceinline__ void tdm_copy_1d_f32(const float* gsrc,
                                                unsigned lds_byte_addr,
                                                unsigned nelem) {
  uint64_t ga = (uint64_t)(uintptr_t)gsrc;
  u32x4 g0;
  g0[0] = 1u;                                        // count=1, user mode
  g0[1] = lds_byte_addr;                             // lds_addr  (bits 63:32)
  g0[2] = (unsigned)(ga & 0xFFFFFFFFu);              // global_addr[31:0]
  g0[3] = (unsigned)((ga >> 32) & 0x1FFFFFFu)        // global_addr[56:32]
        | (2u << 30);                                // type = 2 ("image")
  i32x8 g1;
  g1[0] = (int)(2u << 16);                           // data_size=2 -> 4 bytes, no multicast
  g1[1] = (int)((nelem & 0xFFFFu) << 16);            // tensor_dim0[15:0]  (bits 63:48)
  g1[2] = (int)(((nelem >> 16) & 0xFFFFu)            // tensor_dim0[31:16] (bits 79:64)
        | (1u << 16));                               // tensor_dim1 = 1
  g1[3] = (int)((nelem & 0xFFFFu) << 16);            // tile_dim0 (bits 127:112)
  g1[4] = 1;                                         // tile_dim1 = 1
  g1[5] = (int)nelem;                                // tensor_dim0_stride (low 32)
  g1[6] = 0;
  g1[7] = 0;
  i32x4 z4 = {0, 0, 0, 0};
  i32x8 z8 = {0, 0, 0, 0, 0, 0, 0, 0};
  // 6-arg form (clang-23 / therock-10.0 headers): groups 2,3 + aux, then cpol.
  __builtin_amdgcn_tensor_load_to_lds(g0, g1, z4, z4, z8, 0);
}
#endif

__global__ void PlanarLoss_main(const float* __restrict__ points,
                                const float* __restrict__ normals,
                                const float* __restrict__ offsets,
                                const int*   __restrict__ assign,
                                float* __restrict__ acc,   // acc[0]=sum, acc[1]=count
                                int N) {
  __shared__ float  s_n[NPLANES * 3];
  __shared__ float  s_d[NPLANES];
  __shared__ float4 s_pl[NPLANES];
  __shared__ float  red[2 * BLOCK];

  const int b   = blockIdx.y;
  const int tid = threadIdx.x;

  // ---- Stage the per-batch plane table into LDS (TDM on wave 0) ----
  if (tid < 32) {
#if __has_builtin(__builtin_amdgcn_tensor_load_to_lds)
    tdm_copy_1d_f32(normals + (size_t)b * NPLANES * 3, lds_offset_of(s_n), NPLANES * 3);
    tdm_copy_1d_f32(offsets + (size_t)b * NPLANES,     lds_offset_of(s_d), NPLANES);
    __builtin_amdgcn_s_wait_tensorcnt(0);
#else
    for (int i = tid; i < NPLANES * 3; i += 32) s_n[i] = normals[(size_t)b * NPLANES * 3 + i];
    for (int i = tid; i < NPLANES;     i += 32) s_d[i] = offsets[(size_t)b * NPLANES + i];
#endif
  }
  __syncthreads();
  if (tid < NPLANES)
    s_pl[tid] = make_float4(s_n[3 * tid], s_n[3 * tid + 1], s_n[3 * tid + 2], s_d[tid]);
  __syncthreads();

  // ---- Main gather pass: 4 points / thread, 3x float4 + 1x int4 loads ----
  const float4* pts4 = (const float4*)(points + (size_t)b * N * 3);
  const int4*   asg4 = (const int4*)(assign + (size_t)b * N);
  const int nq = N >> 2;

  float sum = 0.0f;
  float cnt = 0.0f;
  for (int q = blockIdx.x * blockDim.x + tid; q < nq; q += gridDim.x * blockDim.x) {
    float4 P0 = pts4[3 * q + 0];
    float4 P1 = pts4[3 * q + 1];
    float4 P2 = pts4[3 * q + 2];
    int4   A  = asg4[q];
    float px[4] = {P0.x, P0.w, P1.z, P2.y};
    float py[4] = {P0.y, P1.x, P1.w, P2.z};
    float pz[4] = {P0.z, P1.y, P2.x, P2.w};
    int   ai[4] = {A.x, A.y, A.z, A.w};
#pragma unroll
    for (int j = 0; j < 4; ++j) {
      const bool valid = ai[j] >= 0;
      const int  idx   = valid ? ai[j] : 0;
      float4 pl = s_pl[idx];                     // ds_load_b128 gather
      float dist = fabsf(fmaf(px[j], pl.x, fmaf(py[j], pl.y, fmaf(pz[j], pl.z, pl.w))));
      sum += valid ? dist : 0.0f;
      cnt += valid ? 1.0f : 0.0f;
    }
  }
  // Tail (N % 4), handled by block 0 only.
  if (blockIdx.x == 0 && tid < (N & 3)) {
    const int n = (nq << 2) + tid;
    const int a = assign[(size_t)b * N + n];
    if (a >= 0) {
      const float* pp = points + ((size_t)b * N + n) * 3;
      float4 pl = s_pl[a];
      sum += fabsf(fmaf(pp[0], pl.x, fmaf(pp[1], pl.y, fmaf(pp[2], pl.z, pl.w))));
      cnt += 1.0f;
    }
  }

  // ---- Block reduction via v_wmma_f32_16x16x4_f32 (exact f32) ----
  red[tid]         = sum;
  red[BLOCK + tid] = cnt;
  __syncthreads();

  const int wave = tid >> 5;
  const int lane = tid & 31;
  if (wave < 2) {                  // wave0 reduces sums, wave1 reduces counts
    const float* buf = &red[wave * BLOCK];
    v8f c = {};
    const v2f ones = {1.0f, 1.0f};
    // A-layout (32-bit 16x4): lanes 0-15 hold K=0,1; lanes 16-31 hold K=2,3.
    const int r    = (lane < 16) ? lane : (lane - 16);
    const int koff = (lane < 16) ? 0 : 2;
#pragma unroll
    for (int ch = 0; ch < 4; ++ch) {
      const int base = 64 * ch + 4 * r + koff;
      v2f a;
      a[0] = buf[base];
      a[1] = buf[base + 1];
      // D[m][*] += sum_k A[m][k]  -> after 4 chunks, D rows hold chunk row-sums.
      c = __builtin_amdgcn_wmma_f32_16x16x4_f32(false, a, false, ones,
                                                (short)0, c, false, false);
    }
    float s = c[0] + c[1] + c[2] + c[3] + c[4] + c[5] + c[6] + c[7];
    s += __shfl_xor(s, 16);        // combine rows 0-7 with rows 8-15
    if (lane == 0) atomicAdd(&acc[wave], s);
  }
}

__global__ void PlanarLoss_init(float* acc) {
  if (threadIdx.x < 2) acc[threadIdx.x] = 0.0f;
}

__global__ void PlanarLoss_final(const float* __restrict__ acc, float* __restrict__ out) {
  const float t = acc[0];
  const float c = acc[1];
  out[0] = (c > 0.0f) ? (t / c) : t;
}

extern "C" void kernel_launch(void* const* d_in, const int* in_sizes, int n_in,
                              void* d_out, int out_size, void* d_ws, size_t ws_size,
                              hipStream_t stream) {
  const float* points  = (const float*)d_in[0];
  const float* normals = (const float*)d_in[1];
  const float* offsets = (const float*)d_in[2];
  const int*   assign  = (const int*)d_in[3];
  float* out = (float*)d_out;
  float* acc = (float*)d_ws;

  int B = in_sizes[2] / NPLANES;   // plane_offsets is (B, 64)
  if (B <= 0) B = 1;
  int N = in_sizes[3] / B;         // plane_assignments is (B, N)

  PlanarLoss_init<<<1, 32, 0, stream>>>(acc);
  dim3 grid(128, B);
  PlanarLoss_main<<<grid, BLOCK, 0, stream>>>(points, normals, offsets, assign, acc, N);
  PlanarLoss_final<<<1, 1, 0, stream>>>(acc, out);
}